// CannyEdgeFilter_738734375658
// MI455X (gfx1250) — compile-verified
//
#include <hip/hip_runtime.h>
#include <hip/hip_bf16.h>
#include <math.h>

typedef __attribute__((ext_vector_type(2))) float v2f;
typedef __attribute__((ext_vector_type(8))) float v8f;

#define TILE 44   // output tile (per block)
#define XBN  48   // blurred-grid extent  (TILE + 4)
#define GEN  46   // gradient-grid extent (TILE + 2)
#define SXC  52   // staged-input cols (XBN + 4)
#define SXR  64   // staged-input rows, padded to 16 for WMMA row tiles
#define TMPR 64   // horizontal-pass rows (padded)
#define TMPC 48   // horizontal-pass cols

// jnp.pad mode='reflect' index map (valid for our bounded halos)
__device__ __forceinline__ int rfl(int i, int n) {
  if (i < 0) i = -i;
  if (i >= n) i = 2 * n - 2 - i;
  return i;
}

// banded 5-tap Gaussian weight: w[d] for d in [0,4], else 0 (symmetric)
__device__ __forceinline__ float band5(int d, float w0, float w1, float w2) {
  if (d < 0 || d > 4) return 0.f;
  int dd = d > 2 ? 4 - d : d;
  return dd == 0 ? w0 : (dd == 1 ? w1 : w2);
}

__global__ __launch_bounds__(256)
void canny_main(const float* __restrict__ x, float* __restrict__ nmsout,
                float* __restrict__ maxbuf, int H, int W) {
  __shared__ float s_x[SXR][SXC];       // reflect-staged input
  __shared__ float s_tmp[TMPR][TMPC];   // after horizontal gaussian
  __shared__ float s_xb[XBN][XBN];      // after vertical gaussian (blurred)
  __shared__ float s_g[GEN][GEN];       // gradient magnitude (0 outside image)
  __shared__ unsigned char s_bin[GEN][GEN];
  __shared__ float s_red[8];

  const int img = blockIdx.z;
  const int oy = blockIdx.y * TILE;
  const int ox = blockIdx.x * TILE;
  const int Y0 = oy - 2, X0 = ox - 2;   // s_xb[a][b] <-> absolute (Y0+a, X0+b)
  const int tid  = threadIdx.x;
  const int lane = tid & 31;
  const int wid  = tid >> 5;
  const int lr   = lane & 15;           // M (A/C) or N (B/C) index
  const int hi   = lane >> 4;           // half-wave selector

  // Gaussian 1D weights, sigma=1, size=5, computed in f32 like the reference
  const float e1 = expf(-0.5f), e2 = expf(-2.0f);
  const float ws = 1.f + 2.f * e1 + 2.f * e2;
  const float w0 = e2 / ws, w1 = e1 / ws, w2 = 1.f / ws;

  const float* xi = x + (size_t)img * H * W;

  // ---- Stage 0: stage input tile with per-axis reflect indexing ----
  for (int idx = tid; idx < SXR * SXC; idx += 256) {
    int ar = idx / SXC, ac = idx % SXC;
    int ry = rfl(Y0 - 2 + ar, H);
    int rx = rfl(X0 - 2 + ac, W);
    s_x[ar][ac] = xi[(size_t)ry * W + rx];
  }
  __syncthreads();

  // ---- Stage 1: horizontal Gaussian as GEMM (X_ext 16x20 @ band 20x16) ----
  // tmp[ar][b] = sum_t w[t] * s_x[ar][b + t], accumulated via 5x K=4 WMMA.
  for (int t = wid; t < (TMPR / 16) * (TMPC / 16); t += 8) {
    int tr = t / (TMPC / 16), tc = t % (TMPC / 16);
    v8f acc = {};
    for (int q = 0; q < 5; ++q) {
      v2f a, b;
      int row = 16 * tr + lr;
      int col = 16 * tc + 4 * q + 2 * hi;   // A: K = 4q+2*hi (+1 in .y)
      a.x = s_x[row][col];
      a.y = s_x[row][col + 1];
      int k = 4 * q + 2 * hi;               // B: row K = 4q+2*hi (+1 in .y)
      b.x = band5(k - lr,     w0, w1, w2);
      b.y = band5(k + 1 - lr, w0, w1, w2);
      acc = __builtin_amdgcn_wmma_f32_16x16x4_f32(false, a, false, b,
                                                  (short)0, acc, false, false);
    }
    for (int v = 0; v < 8; ++v)             // C/D layout: row = v + 8*hi
      s_tmp[16 * tr + v + 8 * hi][16 * tc + lr] = acc[v];
  }
  __syncthreads();

  // ---- Stage 2: vertical Gaussian as GEMM (band 16x20 @ tmp 20x16) ----
  // s_xb[a][b] = sum_t w[t] * s_tmp[a + t][b]
  for (int t = wid; t < (XBN / 16) * (XBN / 16); t += 8) {
    int vr = t / (XBN / 16), vc = t % (XBN / 16);
    v8f acc = {};
    for (int q = 0; q < 5; ++q) {
      v2f a, b;
      int k = 4 * q + 2 * hi;
      a.x = band5(k - lr,     w0, w1, w2);  // A(M=lr, K=k)
      a.y = band5(k + 1 - lr, w0, w1, w2);
      b.x = s_tmp[16 * vr + k][16 * vc + lr];      // B(K=k, N=lr)
      b.y = s_tmp[16 * vr + k + 1][16 * vc + lr];
      acc = __builtin_amdgcn_wmma_f32_16x16x4_f32(false, a, false, b,
                                                  (short)0, acc, false, false);
    }
    for (int v = 0; v < 8; ++v)
      s_xb[16 * vr + v + 8 * hi][16 * vc + lr] = acc[v];
  }
  __syncthreads();

  // ---- Stage 3: Sobel + magnitude + direction bin (zero outside image) ----
  for (int idx = tid; idx < GEN * GEN; idx += 256) {
    int a2 = idx / GEN, b2 = idx % GEN;
    int p = oy - 1 + a2, qc = ox - 1 + b2;
    float g = 0.f; unsigned char bn = 0;
    if (p >= 0 && p < H && qc >= 0 && qc < W) {
      float xm1m1 = s_xb[a2][b2],     xm10 = s_xb[a2][b2 + 1],     xm1p1 = s_xb[a2][b2 + 2];
      float x0m1  = s_xb[a2 + 1][b2],                              x0p1  = s_xb[a2 + 1][b2 + 2];
      float xp1m1 = s_xb[a2 + 2][b2], xp10 = s_xb[a2 + 2][b2 + 1], xp1p1 = s_xb[a2 + 2][b2 + 2];
      float gx = (xm1p1 - xm1m1) + 2.f * (x0p1 - x0m1) + (xp1p1 - xp1m1);
      float gy = (xm1m1 + 2.f * xm10 + xm1p1) - (xp1m1 + 2.f * xp10 + xp1p1);
      g = sqrtf(gx * gx + gy * gy);
      float ang = atan2f(gy, gx) * 57.29577951308232f;
      if (ang < 0.f) ang += 180.f;
      if (ang < 22.5f || ang >= 157.5f)      bn = 0;
      else if (ang < 67.5f)                  bn = 1;
      else if (ang < 112.5f)                 bn = 2;
      else                                   bn = 3;
    }
    s_g[a2][b2] = g;
    s_bin[a2][b2] = bn;
  }
  __syncthreads();

  // ---- Stage 4: directional NMS + per-image max (scale-free) ----
  const int qdi[4] = {0, 1, 1, -1}, qdj[4] = {1, -1, 0, -1};
  const int rdi[4] = {0, -1, -1, 1}, rdj[4] = {-1, 1, 0, 1};
  float lmax = 0.f;
  for (int idx = tid; idx < TILE * TILE; idx += 256) {
    int r = idx / TILE, c = idx % TILE;
    int p = oy + r, qc = ox + c;
    if (p < H && qc < W) {
      int a2 = r + 1, b2 = c + 1;
      float g = s_g[a2][b2];
      int bn = s_bin[a2][b2];
      float qv = s_g[a2 + qdi[bn]][b2 + qdj[bn]];
      float rv = s_g[a2 + rdi[bn]][b2 + rdj[bn]];
      float nms = (g >= qv && g >= rv) ? g : 0.f;
      nmsout[(size_t)img * H * W + (size_t)p * W + qc] = nms;
      lmax = fmaxf(lmax, nms);
    }
  }
  for (int off = 16; off > 0; off >>= 1)
    lmax = fmaxf(lmax, __shfl_xor(lmax, off, 32));
  if (lane == 0) s_red[wid] = lmax;
  __syncthreads();
  if (tid == 0) {
    float m = s_red[0];
    for (int i = 1; i < 8; ++i) m = fmaxf(m, s_red[i]);
    // nms >= 0, maxbuf init 0 -> IEEE order == int order
    atomicMax((int*)(maxbuf + img), __float_as_int(m));
  }
}

__global__ void canny_init(float* maxbuf, int n) {
  int i = blockIdx.x * blockDim.x + threadIdx.x;
  if (i < n) maxbuf[i] = 0.f;
}

__global__ __launch_bounds__(256)
void canny_thresh(float* __restrict__ out, const float* __restrict__ maxbuf,
                  int n4, int perImg4) {
  int i = blockIdx.x * blockDim.x + threadIdx.x;
  int stride = gridDim.x * blockDim.x;
  float4* o4 = (float4*)out;
  for (; i < n4; i += stride) {
    int img = i / perImg4;
    float m = maxbuf[img];
    float hiT = m * 0.15f, loT = m * 0.05f;
    float4 v = o4[i];
    v.x = v.x >= hiT ? 255.f : (v.x >= loT ? 25.f : 0.f);
    v.y = v.y >= hiT ? 255.f : (v.y >= loT ? 25.f : 0.f);
    v.z = v.z >= hiT ? 255.f : (v.z >= loT ? 25.f : 0.f);
    v.w = v.w >= hiT ? 255.f : (v.w >= loT ? 25.f : 0.f);
    o4[i] = v;
  }
}

extern "C" void kernel_launch(void* const* d_in, const int* in_sizes, int n_in,
                              void* d_out, int out_size, void* d_ws, size_t ws_size,
                              hipStream_t stream) {
  const float* x = (const float*)d_in[0];
  float* out = (float*)d_out;
  const int H = 1024, W = 1024;
  const int B = in_sizes[0] / (H * W);
  float* maxbuf = (float*)d_ws;  // B floats of scratch

  canny_init<<<1, 64, 0, stream>>>(maxbuf, B);

  dim3 grid((W + TILE - 1) / TILE, (H + TILE - 1) / TILE, B);
  canny_main<<<grid, 256, 0, stream>>>(x, out, maxbuf, H, W);

  const int n4 = B * H * W / 4;
  canny_thresh<<<2048, 256, 0, stream>>>(out, maxbuf, n4, H * W / 4);
}